// CustomGATLayer_996432413188
// MI455X (gfx1250) — compile-verified
//
#include <hip/hip_runtime.h>
#include <math.h>

// ---------------------------------------------------------------------------
// GAT layer for MI455X (gfx1250, wave32).
//   N=50000 nodes, E=800000 edges, IN=128, HEADS=4, D=32 (H*D=128)
// Pipeline:
//   0) init: zero d_out, alpha_sum, head-max
//   1) h = x @ W             -> v_wmma_f32_16x16x4_f32, 16x64 strip per wave
//                               (A-fragment reused across 4 accumulators)
//   2) s_src/s_tgt per node  -> tiny L2-resident score tables
//   3) per-head global max   -> ordered-uint atomicMax
//   4) softmax denominator   -> global_atomic_add_f32 into alpha_sum
//   5) weighted scatter      -> wave-per-edge, lane-per-dim, f32 atomics
//   6) batchnorm stats + apply (in place on d_out)
// ---------------------------------------------------------------------------

typedef __attribute__((ext_vector_type(2))) float v2f;
typedef __attribute__((ext_vector_type(8))) float v8f;

#define HD     128   // HEADS*OUT_DIM == IN_DIM
#define HEADS  4
#define ODIM   32
#define LEAKY  0.4f

__device__ __forceinline__ float leaky_relu(float x) {
    return x > 0.0f ? x : LEAKY * x;
}

// monotonic float<->uint encoding for atomicMax on floats
__device__ __forceinline__ unsigned enc_f32(float f) {
    unsigned b = __float_as_uint(f);
    return (b & 0x80000000u) ? ~b : (b | 0x80000000u);
}
__device__ __forceinline__ float dec_f32(unsigned u) {
    unsigned b = (u & 0x80000000u) ? (u & 0x7FFFFFFFu) : ~u;
    return __uint_as_float(b);
}

// native f32 atomic add (lowers to global_atomic_add_f32, no CAS loop)
__device__ __forceinline__ void atom_add_f32(float* p, float v) {
    __hip_atomic_fetch_add(p, v, __ATOMIC_RELAXED, __HIP_MEMORY_SCOPE_AGENT);
}

__device__ __forceinline__ v8f wmma_f32(v2f a, v2f b, v8f c) {
    return __builtin_amdgcn_wmma_f32_16x16x4_f32(
        /*neg_a=*/false, a, /*neg_b=*/false, b,
        /*c_mod=*/(short)0, c, /*reuse_a=*/false, /*reuse_b=*/false);
}

// ---------------------------------------------------------------------------
// 0) init: zero accumulators
// ---------------------------------------------------------------------------
__global__ void gat_init(float* __restrict__ out, float* __restrict__ alpha_sum,
                         unsigned* __restrict__ gmax, int n_out, int n_as) {
    int i = blockIdx.x * blockDim.x + threadIdx.x;
    if (i < n_out) out[i] = 0.0f;
    if (i < n_as)  alpha_sum[i] = 0.0f;
    if (i < HEADS) gmax[i] = 0u;   // 0 < enc(-inf): identity for ordered max
}

// ---------------------------------------------------------------------------
// 1) h = x @ W via V_WMMA_F32_16X16X4_F32.
//    One wave computes a 16x64 strip: 4 accumulators share one A fragment
//    per k-step (A traffic /4, 4 independent WMMA chains for XDL overlap).
//    A 16x4:  lanes 0-15 = rows, half=lane>>4 picks K pair {0,1}/{2,3}
//    B 4x16:  lanes 0-15 = cols, half picks K pair (mirror of A)
//    C/D:     VGPR i holds row m = i + 8*half, col = lane&15
// ---------------------------------------------------------------------------
__global__ void gat_gemm_wmma(const float* __restrict__ x, const float* __restrict__ W,
                              float* __restrict__ h, int N) {
    const int lane = threadIdx.x & 31;
    const int wid  = threadIdx.x >> 5;
    const int MT   = (N + 15) >> 4;                       // row tiles
    const int tile = blockIdx.x * (blockDim.x >> 5) + wid;
    if (tile >= MT * 2) return;                           // wave-uniform

    const int mt   = tile >> 1;       // row tile
    const int ntg  = tile & 1;        // 64-wide column group (128/64 = 2)
    const int half = lane >> 4;
    const int lm   = lane & 15;

    int row = mt * 16 + lm;
    if (row >= N) row = N - 1;        // per-lane clamp; EXEC stays all-ones
    const int col0 = ntg * 64 + lm;
    const float* __restrict__ arow = x + (size_t)row * HD;

    v8f acc0 = {}, acc1 = {}, acc2 = {}, acc3 = {};
#pragma unroll 2
    for (int kk = 0; kk < HD; kk += 4) {
        const int k = kk + 2 * half;
        v2f a = *(const v2f*)(arow + k);                  // A[m][k], A[m][k+1]
        const float* __restrict__ wrow = W + (size_t)k * HD + col0;
        v2f b0, b1, b2, b3;
        b0.x = wrow[0];       b0.y = wrow[HD];
        b1.x = wrow[16];      b1.y = wrow[HD + 16];
        b2.x = wrow[32];      b2.y = wrow[HD + 32];
        b3.x = wrow[48];      b3.y = wrow[HD + 48];
        acc0 = wmma_f32(a, b0, acc0);
        acc1 = wmma_f32(a, b1, acc1);
        acc2 = wmma_f32(a, b2, acc2);
        acc3 = wmma_f32(a, b3, acc3);
    }

    // Epilogue: tile-uniform bounds check -> branchless fast path with
    // immediate-offset stores from a single base pointer.
    float* __restrict__ p = h + (size_t)(mt * 16 + 8 * half) * HD + col0;
    if (mt * 16 + 16 <= N) {
#pragma unroll
        for (int i = 0; i < 8; ++i) {
            p[(size_t)i * HD]      = acc0[i];
            p[(size_t)i * HD + 16] = acc1[i];
            p[(size_t)i * HD + 32] = acc2[i];
            p[(size_t)i * HD + 48] = acc3[i];
        }
    } else {
#pragma unroll
        for (int i = 0; i < 8; ++i) {
            if (mt * 16 + 8 * half + i < N) {
                p[(size_t)i * HD]      = acc0[i];
                p[(size_t)i * HD + 16] = acc1[i];
                p[(size_t)i * HD + 32] = acc2[i];
                p[(size_t)i * HD + 48] = acc3[i];
            }
        }
    }
}

// ---------------------------------------------------------------------------
// 2) per-node attention scores: s_src[n,h] = h[n,h,:].a_src[h], same for tgt
// ---------------------------------------------------------------------------
__global__ void gat_node_scores(const float* __restrict__ h,
                                const float* __restrict__ a_src,
                                const float* __restrict__ a_tgt,
                                float* __restrict__ s_src, float* __restrict__ s_tgt,
                                int N) {
    int idx = blockIdx.x * blockDim.x + threadIdx.x;      // (node, head)
    if (idx >= N * HEADS) return;
    const int n  = idx >> 2;
    const int hd = idx & 3;
    const float* __restrict__ hp = h + (size_t)n * HD + hd * ODIM;
    const float* __restrict__ as = a_src + hd * ODIM;
    const float* __restrict__ at = a_tgt + hd * ODIM;
    float ss = 0.0f, st = 0.0f;
#pragma unroll
    for (int d = 0; d < ODIM; ++d) {
        float v = hp[d];
        ss = fmaf(v, as[d], ss);
        st = fmaf(v, at[d], st);
    }
    s_src[idx] = ss;
    s_tgt[idx] = st;
}

// ---------------------------------------------------------------------------
// 3) per-head global max of leaky_relu(s_src[src]+s_tgt[tgt])
// ---------------------------------------------------------------------------
__global__ void gat_edge_max(const int* __restrict__ src, const int* __restrict__ tgt,
                             const float* __restrict__ s_src, const float* __restrict__ s_tgt,
                             unsigned* __restrict__ gmax, int E) {
    __shared__ unsigned smax[HEADS];
    if (threadIdx.x < HEADS) smax[threadIdx.x] = 0u;
    __syncthreads();
    int e = blockIdx.x * blockDim.x + threadIdx.x;
    if (e < E) {
        const int s = src[e], t = tgt[e];
#pragma unroll
        for (int hd = 0; hd < HEADS; ++hd) {
            float lg = leaky_relu(s_src[s * HEADS + hd] + s_tgt[t * HEADS + hd]);
            atomicMax(&smax[hd], enc_f32(lg));
        }
    }
    __syncthreads();
    if (threadIdx.x < HEADS) atomicMax(&gmax[threadIdx.x], smax[threadIdx.x]);
}

// ---------------------------------------------------------------------------
// 4) softmax denominator: alpha_sum[tgt,h] += exp(logit - max_h)
// ---------------------------------------------------------------------------
__global__ void gat_edge_sum(const int* __restrict__ src, const int* __restrict__ tgt,
                             const float* __restrict__ s_src, const float* __restrict__ s_tgt,
                             const unsigned* __restrict__ gmax,
                             float* __restrict__ alpha_sum, int E) {
    int e = blockIdx.x * blockDim.x + threadIdx.x;
    if (e >= E) return;
    const int s = src[e], t = tgt[e];
#pragma unroll
    for (int hd = 0; hd < HEADS; ++hd) {
        float lg = leaky_relu(s_src[s * HEADS + hd] + s_tgt[t * HEADS + hd]);
        float a  = __expf(lg - dec_f32(gmax[hd]));
        atom_add_f32(&alpha_sum[t * HEADS + hd], a);
    }
}

// ---------------------------------------------------------------------------
// 5) weighted scatter: out[tgt] += alpha * h[src]; one wave32 per edge,
//    lane = dim within head -> coalesced 128B gathers and atomic scatters
// ---------------------------------------------------------------------------
__global__ void gat_scatter(const int* __restrict__ src, const int* __restrict__ tgt,
                            const float* __restrict__ s_src, const float* __restrict__ s_tgt,
                            const unsigned* __restrict__ gmax,
                            const float* __restrict__ alpha_sum,
                            const float* __restrict__ h, float* __restrict__ out, int E) {
    const int lane = threadIdx.x & 31;
    const int e = blockIdx.x * (blockDim.x >> 5) + (threadIdx.x >> 5);
    if (e >= E) return;                                    // wave-uniform
    const int s = src[e], t = tgt[e];
    float w[HEADS];
#pragma unroll
    for (int hd = 0; hd < HEADS; ++hd) {
        float lg = leaky_relu(s_src[s * HEADS + hd] + s_tgt[t * HEADS + hd]);
        float a  = __expf(lg - dec_f32(gmax[hd]));
        w[hd] = a / (alpha_sum[t * HEADS + hd] + 1e-16f);
    }
    const float* __restrict__ hrow = h + (size_t)s * HD;
    float* __restrict__ orow = out + (size_t)t * HD;
#pragma unroll
    for (int hd = 0; hd < HEADS; ++hd) {
        float v = hrow[hd * ODIM + lane] * w[hd];
        atom_add_f32(&orow[hd * ODIM + lane], v);
    }
}

// ---------------------------------------------------------------------------
// 6) batchnorm: per-channel biased mean/var, then in-place normalize
// ---------------------------------------------------------------------------
__global__ void gat_bn_stats(const float* __restrict__ out, float* __restrict__ mean,
                             float* __restrict__ var, int N) {
    __shared__ float ssum[256];
    __shared__ float ssq[256];
    const int c = blockIdx.x;        // 128 channels
    const int t = threadIdx.x;
    float sum = 0.0f, sq = 0.0f;
    for (int r = t; r < N; r += 256) {
        float v = out[(size_t)r * HD + c];
        sum += v;
        sq  = fmaf(v, v, sq);
    }
    ssum[t] = sum; ssq[t] = sq;
    __syncthreads();
    for (int s = 128; s > 0; s >>= 1) {
        if (t < s) { ssum[t] += ssum[t + s]; ssq[t] += ssq[t + s]; }
        __syncthreads();
    }
    if (t == 0) {
        float m = ssum[0] / (float)N;
        mean[c] = m;
        var[c]  = ssq[0] / (float)N - m * m;
    }
}

__global__ void gat_bn_apply(float* __restrict__ out, const float* __restrict__ mean,
                             const float* __restrict__ var, const float* __restrict__ gamma,
                             const float* __restrict__ beta, int total) {
    int i = blockIdx.x * blockDim.x + threadIdx.x;
    if (i >= total) return;
    int c = i & (HD - 1);
    float inv = rsqrtf(var[c] + 1e-5f);
    out[i] = (out[i] - mean[c]) * inv * gamma[c] + beta[c];
}

// ---------------------------------------------------------------------------
// launch
// ---------------------------------------------------------------------------
extern "C" void kernel_launch(void* const* d_in, const int* in_sizes, int n_in,
                              void* d_out, int out_size, void* d_ws, size_t ws_size,
                              hipStream_t stream) {
    const float* x     = (const float*)d_in[0];   // (N,128)
    const float* W     = (const float*)d_in[1];   // (128,128)
    const float* a_src = (const float*)d_in[2];   // (4,32)
    const float* a_tgt = (const float*)d_in[3];   // (4,32)
    const float* gamma = (const float*)d_in[4];   // (128,)
    const float* beta  = (const float*)d_in[5];   // (128,)
    const int*   ei    = (const int*)d_in[6];     // (2,E)

    const int N = in_sizes[0] / HD;
    const int E = in_sizes[6] / 2;
    const int* src = ei;
    const int* tgt = ei + E;
    float* out = (float*)d_out;

    // workspace layout (floats)
    float*    ws        = (float*)d_ws;
    float*    h         = ws;                               // N*128
    float*    s_src     = h + (size_t)N * HD;               // N*4
    float*    s_tgt     = s_src + (size_t)N * HEADS;        // N*4
    float*    alpha_sum = s_tgt + (size_t)N * HEADS;        // N*4
    unsigned* gmax      = (unsigned*)(alpha_sum + (size_t)N * HEADS); // 4
    float*    mean      = (float*)(gmax + HEADS);           // 128
    float*    var       = mean + HD;                        // 128

    const int n_out = N * HD;
    const int n_as  = N * HEADS;

    gat_init<<<(n_out + 255) / 256, 256, 0, stream>>>(out, alpha_sum, gmax, n_out, n_as);

    const int MT    = (N + 15) / 16;
    const int tiles = MT * 2;                 // 16x64 strip per wave
    gat_gemm_wmma<<<(tiles + 7) / 8, 256, 0, stream>>>(x, W, h, N);

    gat_node_scores<<<(N * HEADS + 255) / 256, 256, 0, stream>>>(h, a_src, a_tgt,
                                                                 s_src, s_tgt, N);

    gat_edge_max<<<(E + 255) / 256, 256, 0, stream>>>(src, tgt, s_src, s_tgt, gmax, E);

    gat_edge_sum<<<(E + 255) / 256, 256, 0, stream>>>(src, tgt, s_src, s_tgt, gmax,
                                                      alpha_sum, E);

    gat_scatter<<<(E + 7) / 8, 256, 0, stream>>>(src, tgt, s_src, s_tgt, gmax,
                                                 alpha_sum, h, out, E);

    gat_bn_stats<<<HD, 256, 0, stream>>>(out, mean, var, N);
    gat_bn_apply<<<(n_out + 255) / 256, 256, 0, stream>>>(out, mean, var, gamma, beta, n_out);
}